// HyperNetwork_50534585205260
// MI455X (gfx1250) — compile-verified
//
#include <hip/hip_runtime.h>

// Problem sizes
#define Bn 512
#define Tn 1024
#define In 32
#define Hn 128
#define On 16
#define DHYPn 64
#define P1n 20736     // H*I + H*H + H + H
#define P2n 2064      // O*H + O
#define NTOT 22800    // P1n + P2n, = 16*1425

// d_out region offsets (floats), in reference return order
#define OUT1_OFF  0ull                       // (B,T,H) = 67,108,864
#define HLAST_OFF 67108864ull                // (1,B,H) = 65,536
#define P1_OFF    67174400ull                // (B,P1)  = 10,616,832
#define OUT2_OFF  77791232ull                // (B,T,O) = 8,388,608
#define P2_OFF    86179840ull                // (B,P2)  = 1,056,768

typedef __attribute__((ext_vector_type(2))) float v2f;
typedef __attribute__((ext_vector_type(8))) float v8f;

// ---------------------------------------------------------------------------
// Kernel A: params = hyp_input @ W_hyp^T + b_hyp   (512 x 22800, K=64)
// One wave32 per 16x16 output tile, V_WMMA_F32_16X16X4_F32, 16 K-steps.
// Tiles never straddle the params1/params2 split (P1 % 16 == 0).
// ---------------------------------------------------------------------------
__global__ __launch_bounds__(256) void hyper_gemm(
    const float* __restrict__ hyp,   // (B, 64)
    const float* __restrict__ Wh,    // (22800, 64) row-major
    const float* __restrict__ bh,    // (22800)
    float* out)
{
    const int wave = blockIdx.x * (blockDim.x >> 5) + (threadIdx.x >> 5);
    const int lane = threadIdx.x & 31;
    const int m_t = wave / 1425;          // 32 M-tiles
    const int n_t = wave % 1425;          // 1425 N-tiles
    const int m0 = m_t * 16, n0 = n_t * 16;
    const int lm = lane & 15;
    const int kb = (lane < 16) ? 0 : 2;   // K sub-offset per half-wave
    const int m = m0 + lm;
    const int n = n0 + lm;

    const float bias = bh[n];
    v8f acc;
#pragma unroll
    for (int r = 0; r < 8; ++r) acc[r] = bias;

    const float* arow = hyp + (size_t)m * DHYPn;  // A[m][k]
    const float* brow = Wh  + (size_t)n * DHYPn;  // B[k][n] = W_hyp[n][k]

#pragma unroll
    for (int kk = 0; kk < 16; ++kk) {
        const int k = kk * 4 + kb;
        v2f a; a.x = arow[k]; a.y = arow[k + 1];
        v2f b; b.x = brow[k]; b.y = brow[k + 1];
        acc = __builtin_amdgcn_wmma_f32_16x16x4_f32(
            false, a, false, b, (short)0, acc, false, false);
    }

    const bool inP1 = (n0 < P1n);
#pragma unroll
    for (int r = 0; r < 8; ++r) {
        const int row = m0 + r + ((lane < 16) ? 0 : 8);
        if (inP1)
            out[P1_OFF + (size_t)row * P1n + n] = acc[r];
        else
            out[P2_OFF + (size_t)row * P2n + (n - P1n)] = acc[r];
    }
}

// ---------------------------------------------------------------------------
// Kernel B: sequential RNN scan, one 128-thread block (4 wave32) per sample.
// W_ih (32) and W_hh (128) rows live in VGPRs (reused 1024x).
// x streamed through double-buffered LDS chunks (64 steps / 8 KB each);
// the next chunk's global loads are issued at chunk start and stay
// outstanding behind ~20K VALU-cycles of compute.  h is ping-ponged in LDS
// so each step needs a single barrier.
// ---------------------------------------------------------------------------
#define CHS 64                      // steps per x-chunk
#define NCH (Tn / CHS)              // 16 chunks
#define CHF (CHS * In)              // 2048 floats per chunk

__global__ __launch_bounds__(128) void rnn_scan(
    const float* __restrict__ x,    // (B,T,I)
    const float* __restrict__ h0,   // (1,B,H)
    float* out)
{
    extern __shared__ float sm[];
    // Layout: [0, CHF) xbuf0 | [CHF, 2*CHF) xbuf1 | hbufA[128] | hbufB[128]
    float* hbufA = sm + 2 * CHF;
    float* hbufB = hbufA + Hn;

    const int b = blockIdx.x;
    const int o = threadIdx.x;
    const float* p1b = out + P1_OFF + (size_t)b * P1n;

    // Per-output-row weights -> registers (reused for all 1024 steps)
    float wih[In];
#pragma unroll
    for (int i = 0; i < In; ++i) wih[i] = p1b[o * In + i];
    float whh[Hn];
#pragma unroll
    for (int j = 0; j < Hn; ++j) whh[j] = p1b[Hn * In + o * Hn + j];
    const float bsum = p1b[Hn * In + Hn * Hn + o] + p1b[Hn * In + Hn * Hn + Hn + o];

    const float4* xg4 = (const float4*)(x + (size_t)b * Tn * In);
    float* o1b = out + OUT1_OFF + (size_t)b * Tn * Hn;

    // Preload chunk 0 (each thread: 4 coalesced float4 = 2048 floats total)
    float4 pf[4];
#pragma unroll
    for (int i = 0; i < 4; ++i) pf[i] = xg4[i * 128 + o];
#pragma unroll
    for (int i = 0; i < 4; ++i) ((float4*)sm)[i * 128 + o] = pf[i];

    float* hcur = hbufA;
    float* hnxt = hbufB;
    hcur[o] = h0[(size_t)b * Hn + o];
    __syncthreads();

    for (int c = 0; c < NCH; ++c) {
        // Issue next chunk's global loads now; they complete during compute.
        if (c + 1 < NCH) {
#pragma unroll
            for (int i = 0; i < 4; ++i)
                pf[i] = xg4[(size_t)(c + 1) * (CHF / 4) + i * 128 + o];
        }
        const float* xc = sm + (c & 1) * CHF;

        for (int s = 0; s < CHS; ++s) {
            const float* xs = xc + s * In;
            float acc = bsum;
#pragma unroll
            for (int i4 = 0; i4 < In; i4 += 4) {
                const float4 xv = *(const float4*)&xs[i4];
                acc += wih[i4] * xv.x + wih[i4 + 1] * xv.y
                     + wih[i4 + 2] * xv.z + wih[i4 + 3] * xv.w;
            }
#pragma unroll
            for (int j4 = 0; j4 < Hn; j4 += 4) {
                const float4 hv = *(const float4*)&hcur[j4];
                acc += whh[j4] * hv.x + whh[j4 + 1] * hv.y
                     + whh[j4 + 2] * hv.z + whh[j4 + 3] * hv.w;
            }
            const float hn = tanhf(acc);

            hnxt[o] = hn;                                 // other buffer: no hazard
            o1b[(size_t)(c * CHS + s) * Hn + o] = hn;
            __syncthreads();                              // single barrier per step
            float* tmp = hcur; hcur = hnxt; hnxt = tmp;
        }

        // Commit prefetched chunk into the other buffer, not the one just read.
        if (c + 1 < NCH) {
            float4* dst = (float4*)(sm + ((c + 1) & 1) * CHF);
#pragma unroll
            for (int i = 0; i < 4; ++i) dst[i * 128 + o] = pf[i];
            __syncthreads();
        }
    }

    out[HLAST_OFF + (size_t)b * Hn + o] = hcur[o];
}

// ---------------------------------------------------------------------------
// Kernel C: output2 = output1 @ W_out^T + b_out  per sample.
// M=1024, N=16 (one N-tile), K=128.  Each wave32 caches its B-tile (W_out)
// and bias in registers once, then sweeps 8 M-tiles of 16 rows:
// V_WMMA_F32_16X16X4_F32 x 32 K-steps per tile.
// ---------------------------------------------------------------------------
__global__ __launch_bounds__(256) void out_gemm(float* out)
{
    const int wave = blockIdx.x * (blockDim.x >> 5) + (threadIdx.x >> 5);
    const int lane = threadIdx.x & 31;
    const int b    = wave >> 3;           // 8 waves per sample
    const int grp  = wave & 7;            // each wave: 8 of the 64 M-tiles
    const int lm   = lane & 15;
    const int kb   = (lane < 16) ? 0 : 2;

    const float* Ab  = out + OUT1_OFF + (size_t)b * Tn * Hn;  // (T,H)
    const float* p2b = out + P2_OFF + (size_t)b * P2n;

    // Cache B (W_out^T tile) and bias in registers: reused for 8 M-tiles.
    const float* brow = p2b + (size_t)lm * Hn;     // B[k][n] = W_out[n][k]
    v2f bc[32];
#pragma unroll
    for (int kk = 0; kk < 32; ++kk) {
        const int k = kk * 4 + kb;
        bc[kk].x = brow[k];
        bc[kk].y = brow[k + 1];
    }
    const float bias = p2b[On * Hn + lm];          // b_out[n]

    for (int mm = 0; mm < 8; ++mm) {
        const int m0 = (grp * 8 + mm) * 16;
        v8f acc;
#pragma unroll
        for (int r = 0; r < 8; ++r) acc[r] = bias;

        const float* arow = Ab + (size_t)(m0 + lm) * Hn;   // A[m][k]
#pragma unroll
        for (int kk = 0; kk < 32; ++kk) {
            const int k = kk * 4 + kb;
            v2f a; a.x = arow[k]; a.y = arow[k + 1];
            acc = __builtin_amdgcn_wmma_f32_16x16x4_f32(
                false, a, false, bc[kk], (short)0, acc, false, false);
        }

#pragma unroll
        for (int r = 0; r < 8; ++r) {
            const int row = m0 + r + ((lane < 16) ? 0 : 8);
            out[OUT2_OFF + ((size_t)b * Tn + row) * On + lm] = acc[r];
        }
    }
}

// ---------------------------------------------------------------------------
extern "C" void kernel_launch(void* const* d_in, const int* in_sizes, int n_in,
                              void* d_out, int out_size, void* d_ws, size_t ws_size,
                              hipStream_t stream) {
    const float* hyp = (const float*)d_in[0];  // (B, DHYP)
    const float* x   = (const float*)d_in[1];  // (B, T, I)
    const float* h0  = (const float*)d_in[2];  // (1, B, H)
    const float* Wh  = (const float*)d_in[3];  // (P1+P2, DHYP)
    const float* bh  = (const float*)d_in[4];  // (P1+P2)
    float* out = (float*)d_out;

    // A: 32*1425 = 45600 tiles, 8 waves/block -> 5700 blocks
    hyper_gemm<<<5700, 256, 0, stream>>>(hyp, Wh, bh, out);
    // B: one block per sample; LDS = 2 x-chunks (16KB) + 2 h buffers (1KB)
    rnn_scan<<<Bn, 128, (2 * CHF + 2 * Hn) * sizeof(float), stream>>>(x, h0, out);
    // C: 512 samples * 8 waves = 4096 waves -> 512 blocks
    out_gemm<<<512, 256, 0, stream>>>(out);
}